// GIN_2121713844488
// MI455X (gfx1250) — compile-verified
//
#include <hip/hip_runtime.h>
#include <hip/hip_bf16.h>

#define DIM_H   128
#define DIM_OUT 64

typedef float v2f __attribute__((ext_vector_type(2)));
typedef float v8f __attribute__((ext_vector_type(8)));

// ---------------------------------------------------------------- copy h -> agg
__global__ void copy_f4_kernel(float4* __restrict__ dst, const float4* __restrict__ src, int n4) {
    int i = blockIdx.x * blockDim.x + threadIdx.x;
    if (i < n4) dst[i] = src[i];
}

// ---------------------------------------------------------------- edge scatter-add
// one wave (32 lanes) per edge; each lane moves a float4 chunk of the 128-dim feature
__global__ void scatter_add_kernel(float* __restrict__ agg, const float* __restrict__ h,
                                   const int* __restrict__ edge, int n_edges) {
    int t = blockIdx.x * blockDim.x + threadIdx.x;
    int e = t >> 5;
    if (e >= n_edges) return;
    int part = t & 31;
    int s = edge[e];            // edge_index[0] = src
    int d = edge[n_edges + e];  // edge_index[1] = dst
    const float4 v = *(const float4*)&h[(size_t)s * DIM_H + part * 4];
    float* p = &agg[(size_t)d * DIM_H + part * 4];
    unsafeAtomicAdd(p + 0, v.x);
    unsafeAtomicAdd(p + 1, v.y);
    unsafeAtomicAdd(p + 2, v.z);
    unsafeAtomicAdd(p + 3, v.w);
}

// ---------------------------------------------------------------- WMMA f32 GEMM
// out[M x N] = act(A[M x 128] @ W[128 x N] + bias) (* mask)
// block: 256 threads = 8 waves, 32-row stripe; W staged in LDS in two 64-row phases.
template <int N, bool RELU, bool MASK>
__global__ __launch_bounds__(256) void gemm_kernel(const float* __restrict__ A,
                                                   const float* __restrict__ W,
                                                   const float* __restrict__ bias,
                                                   const float* __restrict__ mask,
                                                   float* __restrict__ out, int M) {
    constexpr int LDA   = DIM_H + 4;     // 132: pad so A-fragment ds_load_b64 is conflict-free
    constexpr int LDW   = N + 4;
    constexpr int NT    = N / 16;        // column tiles
    constexpr int TILES = (2 * NT) / 8;  // row-tiles per wave: 2 (N=128) or 1 (N=64)

    __shared__ float sA[32 * LDA];       // 16.5 KB
    __shared__ float sW[64 * LDW];       // 33 KB (N=128) / 17 KB (N=64)

    const int tid  = threadIdx.x;
    const int lane = tid & 31;
    const int wv   = tid >> 5;
    const int lm   = lane & 15;   // N (B/C) or M (A) index within tile
    const int kh   = lane >> 4;   // K-half for A/B, M-half for C/D
    const int m0   = blockIdx.x * 32;

    // cooperative load of the 32x128 A stripe
#pragma unroll
    for (int i = 0; i < 4; ++i) {
        int f   = tid + i * 256;
        int row = f >> 5;          // 32 float4 per row
        int c4  = (f & 31) * 4;
        float4 v = *(const float4*)&A[(size_t)(m0 + row) * DIM_H + c4];
        *(float4*)&sA[row * LDA + c4] = v;
    }

    const int ct  = (TILES == 2) ? wv : (wv % NT);
    const int rt0 = (TILES == 2) ? 0 : (wv / NT);

    v8f acc0 = {0.f, 0.f, 0.f, 0.f, 0.f, 0.f, 0.f, 0.f};
    v8f acc1 = {0.f, 0.f, 0.f, 0.f, 0.f, 0.f, 0.f, 0.f};

#pragma unroll
    for (int phase = 0; phase < 2; ++phase) {
        const int k0 = phase * 64;
        __syncthreads();
        constexpr int NW4 = N / 4;
#pragma unroll
        for (int i = 0; i < (64 * NW4) / 256; ++i) {
            int f   = tid + i * 256;
            int row = f / NW4;
            int c4  = (f % NW4) * 4;
            float4 v = *(const float4*)&W[(size_t)(k0 + row) * N + c4];
            *(float4*)&sW[row * LDW + c4] = v;
        }
        __syncthreads();
#pragma unroll
        for (int kk = 0; kk < 64; kk += 4) {
            // B fragment: 4x16, lane = n, lane-half = K-half, vgpr j = K within half
            v2f bf;
            bf.x = sW[(kk + kh * 2 + 0) * LDW + ct * 16 + lm];
            bf.y = sW[(kk + kh * 2 + 1) * LDW + ct * 16 + lm];
            // A fragment: 16x4, lane = m, lane-half = K-half
            v2f af0 = *(const v2f*)&sA[(rt0 * 16 + lm) * LDA + k0 + kk + kh * 2];
            acc0 = __builtin_amdgcn_wmma_f32_16x16x4_f32(false, af0, false, bf,
                                                         (short)0, acc0, false, false);
            if (TILES == 2) {
                v2f af1 = *(const v2f*)&sA[(16 + lm) * LDA + k0 + kk + kh * 2];
                acc1 = __builtin_amdgcn_wmma_f32_16x16x4_f32(false, af1, false, bf,
                                                             (short)0, acc1, false, false);
            }
        }
    }

    // epilogue: C/D layout — lane = n, row = 8*(lane>>4) + vgpr j
    const float bv = bias[ct * 16 + lm];
#pragma unroll
    for (int j = 0; j < 8; ++j) {
        int row = m0 + rt0 * 16 + kh * 8 + j;
        int col = ct * 16 + lm;
        float v = acc0[j] + bv;
        if (RELU) v = fmaxf(v, 0.f);
        if (MASK) v *= mask[(size_t)row * DIM_H + col];
        out[(size_t)row * N + col] = v;
    }
    if (TILES == 2) {
#pragma unroll
        for (int j = 0; j < 8; ++j) {
            int row = m0 + 16 + kh * 8 + j;
            int col = ct * 16 + lm;
            float v = acc1[j] + bv;
            if (RELU) v = fmaxf(v, 0.f);
            if (MASK) v *= mask[(size_t)row * DIM_H + col];
            out[(size_t)row * N + col] = v;
        }
    }
}

// ---------------------------------------------------------------- batchnorm pieces
__global__ void zero_kernel(float* p, int n) {
    int i = blockIdx.x * blockDim.x + threadIdx.x;
    if (i < n) p[i] = 0.f;
}

__global__ void bn_reduce_kernel(const float* __restrict__ z, float* __restrict__ stats, int M) {
    int c  = threadIdx.x & 127;
    int r0 = blockIdx.x * 2 + (threadIdx.x >> 7);
    int stride = gridDim.x * 2;
    float s = 0.f, ss = 0.f;
    for (int r = r0; r < M; r += stride) {
        float v = z[(size_t)r * DIM_H + c];
        s += v;
        ss += v * v;
    }
    unsafeAtomicAdd(&stats[c], s);
    unsafeAtomicAdd(&stats[DIM_H + c], ss);
}

__global__ void bn_finalize_kernel(const float* __restrict__ stats,
                                   const float* __restrict__ gamma,
                                   const float* __restrict__ beta,
                                   float* __restrict__ ss, int M) {
    int c = threadIdx.x;
    float inv  = 1.0f / (float)M;
    float mean = stats[c] * inv;
    float var  = stats[DIM_H + c] * inv - mean * mean;
    float sc   = gamma[c] * rsqrtf(var + 1e-5f);
    ss[c]         = sc;
    ss[DIM_H + c] = beta[c] - mean * sc;
}

__global__ void bn_apply_kernel(const float* __restrict__ z, const float* __restrict__ ss,
                                float* __restrict__ h, int n4) {
    int i = blockIdx.x * blockDim.x + threadIdx.x;
    if (i >= n4) return;
    int c4   = (i & 31) * 4;   // 32 float4 per 128-wide row
    float4 v = *(const float4*)&z[(size_t)i * 4];
    float4 o;
    o.x = v.x * ss[c4 + 0] + ss[DIM_H + c4 + 0];
    o.y = v.y * ss[c4 + 1] + ss[DIM_H + c4 + 1];
    o.z = v.z * ss[c4 + 2] + ss[DIM_H + c4 + 2];
    o.w = v.w * ss[c4 + 3] + ss[DIM_H + c4 + 3];
    *(float4*)&h[(size_t)i * 4] = o;
}

// ---------------------------------------------------------------- row log_softmax (64 cols)
__global__ void log_softmax_kernel(const float* __restrict__ logits, float* __restrict__ out,
                                   int rows) {
    int row = blockIdx.x * 8 + (threadIdx.x >> 5);
    if (row >= rows) return;
    int lane = threadIdx.x & 31;
    float2 v = *(const float2*)&logits[(size_t)row * DIM_OUT + lane * 2];
    float m = fmaxf(v.x, v.y);
#pragma unroll
    for (int d = 16; d >= 1; d >>= 1) m = fmaxf(m, __shfl_xor(m, d, 32));
    float s = __expf(v.x - m) + __expf(v.y - m);
#pragma unroll
    for (int d = 16; d >= 1; d >>= 1) s += __shfl_xor(s, d, 32);
    float lse = m + __logf(s);
    float2 o;
    o.x = v.x - lse;
    o.y = v.y - lse;
    *(float2*)&out[(size_t)row * DIM_OUT + lane * 2] = o;
}

// ---------------------------------------------------------------- driver
extern "C" void kernel_launch(void* const* d_in, const int* in_sizes, int n_in,
                              void* d_out, int out_size, void* d_ws, size_t ws_size,
                              hipStream_t stream) {
    const float* x        = (const float*)d_in[0];
    const int*   edge     = (const int*)d_in[1];
    const float* conv_w1  = (const float*)d_in[2];
    const float* conv_b1  = (const float*)d_in[3];
    const float* conv_w2  = (const float*)d_in[4];
    const float* conv_b2  = (const float*)d_in[5];
    const float* bn_gamma = (const float*)d_in[6];
    const float* bn_beta  = (const float*)d_in[7];
    const float* fc1_w    = (const float*)d_in[8];
    const float* fc1_b    = (const float*)d_in[9];
    const float* fc2_w    = (const float*)d_in[10];
    const float* fc2_b    = (const float*)d_in[11];
    const float* drop     = (const float*)d_in[12];

    const int M = in_sizes[0] / DIM_H;  // 100000 (multiple of 32)
    const int E = in_sizes[1] / 2;      // 600000
    const size_t feat = (size_t)M * DIM_H;

    float* AGG   = (float*)d_ws;
    float* H     = AGG + feat;
    float* T     = H + feat;
    float* STATS = T + feat;    // 256 floats: sum | sumsq
    float* SS    = STATS + 256; // 256 floats: scale | shift

    const int n4            = (int)(feat / 4);
    const int copyBlocks    = (n4 + 255) / 256;
    const int scatterBlocks = (E * 32 + 255) / 256;
    const int gemmBlocks    = M / 32;

    const float* cur = x;
    for (int l = 0; l < 3; ++l) {
        // agg = h + segment_sum(h[src], dst)
        copy_f4_kernel<<<copyBlocks, 256, 0, stream>>>((float4*)AGG, (const float4*)cur, n4);
        scatter_add_kernel<<<scatterBlocks, 256, 0, stream>>>(AGG, cur, edge, E);
        // z = ReLU(ReLU(agg @ W1 + b1) @ W2 + b2)
        gemm_kernel<128, true, false><<<gemmBlocks, 256, 0, stream>>>(
            AGG, conv_w1 + (size_t)l * DIM_H * DIM_H, conv_b1 + l * DIM_H, nullptr, T, M);
        gemm_kernel<128, true, false><<<gemmBlocks, 256, 0, stream>>>(
            T, conv_w2 + (size_t)l * DIM_H * DIM_H, conv_b2 + l * DIM_H, nullptr, AGG, M);
        // batchnorm (training-mode batch stats, biased variance)
        zero_kernel<<<1, 256, 0, stream>>>(STATS, 256);
        bn_reduce_kernel<<<1024, 256, 0, stream>>>(AGG, STATS, M);
        bn_finalize_kernel<<<1, 128, 0, stream>>>(STATS, bn_gamma + l * DIM_H,
                                                  bn_beta + l * DIM_H, SS, M);
        bn_apply_kernel<<<copyBlocks, 256, 0, stream>>>(AGG, SS, H, n4);
        cur = H;
    }
    // head: ReLU(h @ fc1 + b) * dropout_mask, then fc2, then log_softmax
    gemm_kernel<128, true, true><<<gemmBlocks, 256, 0, stream>>>(H, fc1_w, fc1_b, drop, AGG, M);
    gemm_kernel<64, false, false><<<gemmBlocks, 256, 0, stream>>>(AGG, fc2_w, fc2_b, nullptr, T, M);
    log_softmax_kernel<<<(M + 7) / 8, 256, 0, stream>>>(T, (float*)d_out, M);
}